// Receiver_18743237280010
// MI455X (gfx1250) — compile-verified
//
#include <hip/hip_runtime.h>
#include <math.h>

typedef __attribute__((ext_vector_type(2))) float v2f;
typedef __attribute__((ext_vector_type(8))) float v8f;

#define HC 128          // H*C
#define HEADS 2
#define KDIM 128        // inner dim of conv2 GEMMs / fc

// ---------- ordered-float encoding for atomic max on f32 ----------
__device__ __forceinline__ unsigned encF(float f) {
    unsigned u = __float_as_uint(f);
    return (u & 0x80000000u) ? ~u : (u | 0x80000000u);
}
__device__ __forceinline__ float decF(unsigned u) {
    return (u & 0x80000000u) ? __uint_as_float(u & 0x7FFFFFFFu)
                             : __uint_as_float(~u);
}
#define ENC_NEG_INF 0x00800000u   // encF(-FLT_MAX)

// ---------- generic u32 fill ----------
__global__ void fill_u32(unsigned* p, unsigned v, int n) {
    int i = blockIdx.x * blockDim.x + threadIdx.x;
    if (i < n) p[i] = v;
}

// ---------- out2 init: out2[n][k] = b2[k] ----------
__global__ void out_init_bias(float* out, const float* b, int total) {
    int i = blockIdx.x * blockDim.x + threadIdx.x;
    if (i < total) out[i] = b[i & (HC - 1)];
}

// ---------- msgT[k][m] = (message @ Wfc + bfc)^T ----------
__global__ void msgT_kernel(const float* msg, const float* Wfc, const float* bfc,
                            float* msgT, int M) {
    int i = blockIdx.x * blockDim.x + threadIdx.x;
    if (i >= KDIM * M) return;
    int k = i / M, m = i % M;
    float acc = bfc[k];
    for (int j = 0; j < KDIM; ++j) acc += msg[m * KDIM + j] * Wfc[j * HC + k];
    msgT[k * M + m] = acc;
}

// ---------- conv1: per-edge alpha + segment max (rank-1 features) ----------
__global__ void conv1_alpha(const float* x, const float* ea,
                            const int* src, const int* dst,
                            const float* Wl, const float* bl,
                            const float* Wr, const float* br,
                            const float* We, const float* att,
                            float* alphaE, unsigned* amaxEnc, int E) {
    __shared__ float sWl[HC], sWr[HC], sWe[HC], sAtt[HC], sB[HC];
    int t = threadIdx.x;
    if (t < HC) {
        sWl[t] = Wl[t]; sWr[t] = Wr[t]; sWe[t] = We[t];
        sAtt[t] = att[t]; sB[t] = bl[t] + br[t];
    }
    __syncthreads();
    int e = blockIdx.x * blockDim.x + t;
    if (e >= E) return;
    float xs = x[src[e]], xd = x[dst[e]], w = ea[e];
    float a0 = 0.f, a1 = 0.f;
    #pragma unroll 8
    for (int k = 0; k < HC; ++k) {
        float m = fmaf(xs, sWl[k], fmaf(xd, sWr[k], fmaf(w, sWe[k], sB[k])));
        m = (m > 0.f) ? m : 0.2f * m;
        if (k < 64) a0 += m * sAtt[k]; else a1 += m * sAtt[k];
    }
    int d = dst[e];
    alphaE[e * 2 + 0] = a0;
    alphaE[e * 2 + 1] = a1;
    atomicMax(&amaxEnc[d * 2 + 0], encF(a0));
    atomicMax(&amaxEnc[d * 2 + 1], encF(a1));
}

// ---------- shared by both convs: ex = exp(alpha - amax[dst]); denom += ex ----------
__global__ void edge_exp(float* alphaE, const unsigned* amaxEnc, float* denom,
                         const int* dst, int E) {
    int e = blockIdx.x * blockDim.x + threadIdx.x;
    if (e >= E) return;
    int d = dst[e];
    #pragma unroll
    for (int h = 0; h < HEADS; ++h) {
        float ex = __expf(alphaE[e * 2 + h] - decF(amaxEnc[d * 2 + h]));
        alphaE[e * 2 + h] = ex;
        atomicAdd(&denom[d * 2 + h], ex);
    }
}

// ---------- conv1: accumulate S0 = sum a, S1 = sum a*x_src ----------
__global__ void conv1_accum(const float* x, const float* alphaE, const float* denom,
                            const int* src, const int* dst,
                            float* S0, float* S1, int E) {
    int e = blockIdx.x * blockDim.x + threadIdx.x;
    if (e >= E) return;
    float xs = x[src[e]];
    int d = dst[e];
    #pragma unroll
    for (int h = 0; h < HEADS; ++h) {
        float a = alphaE[e * 2 + h] / (denom[d * 2 + h] + 1e-16f);
        atomicAdd(&S0[d * 2 + h], a);
        atomicAdd(&S1[d * 2 + h], a * xs);
    }
}

// ---------- conv1: h1[n][k] = relu(S1*Wl1[k] + S0*bl1[k] + b1[k]) ----------
__global__ void conv1_node(const float* S0, const float* S1,
                           const float* Wl, const float* bl, const float* b1,
                           float* h1, int N) {
    int i = blockIdx.x * blockDim.x + threadIdx.x;   // one thread = 4 channels
    if (i >= N * (HC / 4)) return;
    int n = i >> 5, k0 = (i & 31) * 4;
    int h = k0 >> 6;
    float s0 = S0[n * 2 + h], s1 = S1[n * 2 + h];
    #pragma unroll
    for (int j = 0; j < 4; ++j) {
        int k = k0 + j;
        float v = fmaf(s1, Wl[k], fmaf(s0, bl[k], b1[k]));
        h1[n * HC + k] = fmaxf(v, 0.f);
    }
}

// ---------- WMMA f32 GEMM: out[nrows,ncols] = A[nrows,128] @ W[128,ncols] + bias ----------
__global__ void gemm_wmma_f32(const float* __restrict__ A, const float* __restrict__ W,
                              const float* __restrict__ bias, float* __restrict__ out,
                              int nrows, int ncols) {
    int lane = threadIdx.x & 31;
    int wave = threadIdx.x >> 5;
    int tilesN = ncols >> 4;
    int tile = blockIdx.x * (blockDim.x >> 5) + wave;
    int tilesTotal = (nrows >> 4) * tilesN;
    if (tile >= tilesTotal) return;
    int tm = tile / tilesN, tn = tile % tilesN;
    int half = lane >> 4, l16 = lane & 15;
    int row = tm * 16 + l16;
    int colBase = tn * 16;

    v8f acc = {};
    const float* Arow = A + (size_t)row * KDIM;
    for (int k = 0; k < KDIM; k += 4) {
        int ka = k + 2 * half;
        v2f a, b;
        a.x = Arow[ka];
        a.y = Arow[ka + 1];
        b.x = W[(size_t)ka * ncols + colBase + l16];
        b.y = W[(size_t)(ka + 1) * ncols + colBase + l16];
        acc = __builtin_amdgcn_wmma_f32_16x16x4_f32(
            /*neg_a=*/false, a, /*neg_b=*/false, b,
            /*c_mod=*/(short)0, acc, /*reuse_a=*/false, /*reuse_b=*/false);
    }
    int col = colBase + l16;
    float bv = bias ? bias[col] : 0.f;
    #pragma unroll
    for (int j = 0; j < 8; ++j) {
        int r = tm * 16 + j + 8 * half;
        out[(size_t)r * ncols + col] = acc[j] + bv;
    }
}

// ---------- conv2: wave-cooperative alpha + segment max ----------
__global__ void conv2_alpha(const float* __restrict__ xl2, const float* __restrict__ xr2,
                            const float* __restrict__ ea,
                            const int* __restrict__ src, const int* __restrict__ dst,
                            const float* __restrict__ We, const float* __restrict__ att,
                            float* alphaE, unsigned* amaxEnc, int E) {
    int lane = threadIdx.x & 31;
    int wave = blockIdx.x * (blockDim.x >> 5) + (threadIdx.x >> 5);
    int nWaves = gridDim.x * (blockDim.x >> 5);
    int c0 = lane * 4;
    float4 we = *(const float4*)(We + c0);
    float4 at = *(const float4*)(att + c0);
    int half = lane >> 4, l16 = lane & 15;

    for (int e = wave; e < E; e += nWaves) {
        int s = src[e], d = dst[e];
        float w = ea[e];
        float4 L = *(const float4*)(xl2 + (size_t)s * HC + c0);
        float4 R = *(const float4*)(xr2 + (size_t)d * HC + c0);
        float m, acc = 0.f;
        m = L.x + R.x + w * we.x; m = (m > 0.f) ? m : 0.2f * m; acc += m * at.x;
        m = L.y + R.y + w * we.y; m = (m > 0.f) ? m : 0.2f * m; acc += m * at.y;
        m = L.z + R.z + w * we.z; m = (m > 0.f) ? m : 0.2f * m; acc += m * at.z;
        m = L.w + R.w + w * we.w; m = (m > 0.f) ? m : 0.2f * m; acc += m * at.w;
        // reduce within each 16-lane half (head 0 = lanes 0-15, head 1 = 16-31)
        acc += __shfl_xor(acc, 8, 32);
        acc += __shfl_xor(acc, 4, 32);
        acc += __shfl_xor(acc, 2, 32);
        acc += __shfl_xor(acc, 1, 32);
        if (l16 == 0) {
            alphaE[e * 2 + half] = acc;
            atomicMax(&amaxEnc[d * 2 + half], encF(acc));
        }
    }
}

// ---------- conv2: wave-cooperative weighted scatter of xl2[src] ----------
__global__ void conv2_scatter(const float* __restrict__ xl2,
                              const float* __restrict__ alphaE, const float* __restrict__ denom,
                              const int* __restrict__ src, const int* __restrict__ dst,
                              float* out2, int E) {
    int lane = threadIdx.x & 31;
    int wave = blockIdx.x * (blockDim.x >> 5) + (threadIdx.x >> 5);
    int nWaves = gridDim.x * (blockDim.x >> 5);
    int c0 = lane * 4;
    int half = lane >> 4;

    for (int e = wave; e < E; e += nWaves) {
        int s = src[e], d = dst[e];
        float a = alphaE[e * 2 + half] / (denom[d * 2 + half] + 1e-16f);
        float4 L = *(const float4*)(xl2 + (size_t)s * HC + c0);
        float* o = out2 + (size_t)d * HC + c0;
        atomicAdd(o + 0, a * L.x);
        atomicAdd(o + 1, a * L.y);
        atomicAdd(o + 2, a * L.z);
        atomicAdd(o + 3, a * L.w);
    }
}

// ---------- softmax over nodes (axis 0), M=32 columns ----------
__global__ void sm_colmax(const float* dots, unsigned* colmaxEnc, int N) {
    __shared__ float red[256];
    int col = threadIdx.x & 31, rsub = threadIdx.x >> 5;
    float mx = -3.402823466e+38f;
    int r0 = blockIdx.x * 64;
    for (int r = r0 + rsub; r < r0 + 64 && r < N; r += 8)
        mx = fmaxf(mx, dots[(size_t)r * 32 + col]);
    red[threadIdx.x] = mx;
    __syncthreads();
    if (threadIdx.x < 32) {
        float m = red[threadIdx.x];
        #pragma unroll
        for (int j = 1; j < 8; ++j) m = fmaxf(m, red[threadIdx.x + 32 * j]);
        atomicMax(&colmaxEnc[threadIdx.x], encF(m));
    }
}

__global__ void sm_expsum(float* dots, const unsigned* colmaxEnc, float* colsum, int N) {
    __shared__ float red[256];
    int col = threadIdx.x & 31, rsub = threadIdx.x >> 5;
    float mx = decF(colmaxEnc[col]);
    float s = 0.f;
    int r0 = blockIdx.x * 64;
    for (int r = r0 + rsub; r < r0 + 64 && r < N; r += 8) {
        size_t i = (size_t)r * 32 + col;
        float e = __expf(dots[i] - mx);
        dots[i] = e;
        s += e;
    }
    red[threadIdx.x] = s;
    __syncthreads();
    if (threadIdx.x < 32) {
        float t = red[threadIdx.x];
        #pragma unroll
        for (int j = 1; j < 8; ++j) t += red[threadIdx.x + 32 * j];
        atomicAdd(&colsum[threadIdx.x], t);
    }
}

__global__ void sm_norm(float* dots, const float* colsum, int total) {
    int i = blockIdx.x * blockDim.x + threadIdx.x;
    if (i < total) dots[i] = dots[i] / colsum[i & 31];
}

// =====================================================================
extern "C" void kernel_launch(void* const* d_in, const int* in_sizes, int n_in,
                              void* d_out, int out_size, void* d_ws, size_t ws_size,
                              hipStream_t stream) {
    const int N = in_sizes[0];
    const int E = in_sizes[1] / 2;
    const int M = in_sizes[3] / KDIM;

    const float* x    = (const float*)d_in[0];
    const int*   eidx = (const int*)d_in[1];
    const float* ea   = (const float*)d_in[2];
    const float* msg  = (const float*)d_in[3];
    const float* Wl1  = (const float*)d_in[4];
    const float* bl1  = (const float*)d_in[5];
    const float* Wr1  = (const float*)d_in[6];
    const float* br1  = (const float*)d_in[7];
    const float* We1  = (const float*)d_in[8];
    const float* att1 = (const float*)d_in[9];
    const float* b1   = (const float*)d_in[10];
    const float* Wl2  = (const float*)d_in[11];
    const float* bl2  = (const float*)d_in[12];
    const float* Wr2  = (const float*)d_in[13];
    const float* br2  = (const float*)d_in[14];
    const float* We2  = (const float*)d_in[15];
    const float* att2 = (const float*)d_in[16];
    const float* b2   = (const float*)d_in[17];
    const float* Wfc  = (const float*)d_in[18];
    const float* bfc  = (const float*)d_in[19];

    const int* src = eidx;        // edge_index[0]
    const int* dst = eidx + E;    // edge_index[1]

    // ---- workspace carve-up (256B aligned); out2 aliases h1 ----
    char* ws = (char*)d_ws;
    size_t off = 0;
    auto carve = [&](size_t bytes) { size_t c = off; off = (off + bytes + 255) & ~(size_t)255; return c; };
    float*    h1       = (float*)(ws + carve((size_t)N * HC * 4));   // also out2
    float*    xl2      = (float*)(ws + carve((size_t)N * HC * 4));
    float*    xr2      = (float*)(ws + carve((size_t)N * HC * 4));
    float*    alphaE   = (float*)(ws + carve((size_t)E * 2 * 4));
    unsigned* amaxEnc  = (unsigned*)(ws + carve((size_t)N * 2 * 4));
    float*    denom    = (float*)(ws + carve((size_t)N * 2 * 4));
    float*    S0       = (float*)(ws + carve((size_t)N * 2 * 4));
    float*    S1       = (float*)(ws + carve((size_t)N * 2 * 4));
    float*    msgT     = (float*)(ws + carve((size_t)KDIM * M * 4));
    unsigned* colmax   = (unsigned*)(ws + carve(32 * 4));
    float*    colsum   = (float*)(ws + carve(32 * 4));
    float*    out2     = h1;
    float*    dots     = (float*)d_out;
    (void)ws_size; (void)n_in; (void)out_size;

    const int T = 256;
    auto cdiv = [](int a, int b) { return (a + b - 1) / b; };

    // ---- conv1 ----
    fill_u32<<<cdiv(N * 2, T), T, 0, stream>>>(amaxEnc, ENC_NEG_INF, N * 2);
    fill_u32<<<cdiv(N * 2, T), T, 0, stream>>>((unsigned*)denom, 0u, N * 2);
    fill_u32<<<cdiv(N * 2, T), T, 0, stream>>>((unsigned*)S0, 0u, N * 2);
    fill_u32<<<cdiv(N * 2, T), T, 0, stream>>>((unsigned*)S1, 0u, N * 2);
    msgT_kernel<<<cdiv(KDIM * M, T), T, 0, stream>>>(msg, Wfc, bfc, msgT, M);

    conv1_alpha<<<cdiv(E, T), T, 0, stream>>>(x, ea, src, dst, Wl1, bl1, Wr1, br1,
                                              We1, att1, alphaE, amaxEnc, E);
    edge_exp<<<cdiv(E, T), T, 0, stream>>>(alphaE, amaxEnc, denom, dst, E);
    conv1_accum<<<cdiv(E, T), T, 0, stream>>>(x, alphaE, denom, src, dst, S0, S1, E);
    conv1_node<<<cdiv(N * (HC / 4), T), T, 0, stream>>>(S0, S1, Wl1, bl1, b1, h1, N);

    // ---- conv2 node transforms: WMMA GEMMs ----
    {
        int tiles = (N / 16) * (HC / 16);
        gemm_wmma_f32<<<cdiv(tiles, 8), T, 0, stream>>>(h1, Wl2, bl2, xl2, N, HC);
        gemm_wmma_f32<<<cdiv(tiles, 8), T, 0, stream>>>(h1, Wr2, br2, xr2, N, HC);
    }

    // ---- conv2 edge phase (out2 aliases h1; prior GEMMs already consumed h1) ----
    fill_u32<<<cdiv(N * 2, T), T, 0, stream>>>(amaxEnc, ENC_NEG_INF, N * 2);
    fill_u32<<<cdiv(N * 2, T), T, 0, stream>>>((unsigned*)denom, 0u, N * 2);
    out_init_bias<<<cdiv(N * HC, T), T, 0, stream>>>(out2, b2, N * HC);

    conv2_alpha<<<4096, T, 0, stream>>>(xl2, xr2, ea, src, dst, We2, att2,
                                        alphaE, amaxEnc, E);
    edge_exp<<<cdiv(E, T), T, 0, stream>>>(alphaE, amaxEnc, denom, dst, E);
    conv2_scatter<<<4096, T, 0, stream>>>(xl2, alphaE, denom, src, dst, out2, E);

    // ---- dots = h2 @ msg_emb^T via WMMA ----
    {
        int tiles = (N / 16) * (M / 16);
        gemm_wmma_f32<<<cdiv(tiles, 8), T, 0, stream>>>(out2, msgT, nullptr, dots, N, M);
    }

    // ---- softmax over node axis ----
    fill_u32<<<1, 64, 0, stream>>>(colmax, ENC_NEG_INF, 32);
    fill_u32<<<1, 64, 0, stream>>>((unsigned*)colsum, 0u, 32);
    sm_colmax<<<cdiv(N, 64), T, 0, stream>>>(dots, colmax, N);
    sm_expsum<<<cdiv(N, 64), T, 0, stream>>>(dots, colmax, colsum, N);
    sm_norm<<<cdiv(N * M, T), T, 0, stream>>>(dots, colsum, N * M);
}